// decision_node_53858889892406
// MI455X (gfx1250) — compile-verified
//
#include <hip/hip_runtime.h>
#include <math.h>

typedef float v2f __attribute__((ext_vector_type(2)));
typedef float v8f __attribute__((ext_vector_type(8)));
typedef int   v4i __attribute__((ext_vector_type(4)));

#define GLOBAL_AS __attribute__((address_space(1)))
#define LDS_AS    __attribute__((address_space(3)))

#if __has_builtin(__builtin_amdgcn_global_load_async_to_lds_b128)
#define HAS_ASYNC_LDS 1
#define MIN_BLOCKS 2
#else
#define HAS_ASYNC_LDS 0
#define MIN_BLOCKS 1   // staging regs need >128 VGPRs; don't force 2 blocks/WGP
#endif

namespace {
constexpr int kTrees   = 64;
constexpr int kInF     = 16080;
constexpr int kLeaves  = 128;
constexpr int kClasses = 16;
constexpr int kM       = 4096;              // B*T GEMM rows
constexpr int kKChunk  = 240;               // 16080 = 67 * 240
constexpr int kNChunks = kInF / kKChunk;    // 67
constexpr int kLda     = 244;               // padded LDS row stride -> conflict-free b64 A reads
constexpr int kMTiles  = 2;                 // two 16-row M tiles per wave (2 WMMA chains)
constexpr int kMpb     = kMTiles * 16;      // 32 rows per block
constexpr int kThreads = 256;               // 8 waves; wave w owns leaves w*16..w*16+15
constexpr int kV4      = kMpb * kKChunk / 4;       // 1920 b128 transfers per chunk
constexpr int kV4Tail  = kV4 - 7 * kThreads;       // 128 threads active in slot 7
}

__device__ __forceinline__ void wait_async_zero() {
#if __has_builtin(__builtin_amdgcn_s_wait_asynccnt)
  __builtin_amdgcn_s_wait_asynccnt(0);
#else
  asm volatile("s_wait_asynccnt 0x0" ::: "memory");
#endif
}

__launch_bounds__(kThreads, MIN_BLOCKS)
__global__ void forest_fused_kernel(const float* __restrict__ x,
                                    const float* __restrict__ Wm,
                                    const float* __restrict__ bias,
                                    const float* __restrict__ con,
                                    float* __restrict__ out)
{
  __shared__ float As[2][kMpb * kLda];      // 2 x 31,232 B double buffer for the x tile

  const int tid   = threadIdx.x;
  const int lane  = tid & 31;
  const int wv    = tid >> 5;               // N tile index 0..7
  const int mbase = blockIdx.x * kMpb;

  // WMMA f32 16x16x4 fragment mapping (ISA 7.12.2):
  //   A (16x4, MxK): lanes 0-15 hold row M=lane, K={k0,k0+1}; lanes 16-31 hold K={k0+2,k0+3}
  //   B (4x16, KxN): lanes 0-15 hold col N=lane, K={k0,k0+1}; lanes 16-31 hold K={k0+2,k0+3}
  const int n    = lane & 15;
  const int koff = (lane >> 4) << 1;        // 0 or 2

  const float* wrow = Wm + (size_t)(wv * 16 + n) * kInF + koff;  // B source (L2-resident)
  const float* xblk = x + (size_t)mbase * kInF;

  v8f acc0 = {};
  v8f acc1 = {};

  // Thread-invariant copy-slot addressing (hoisted out of the chunk loop).
  int goffs[8], loffs[8];
  #pragma unroll
  for (int j = 0; j < 8; ++j) {
    const int idx = j * kThreads + tid;
    const int r   = idx / (kKChunk / 4);
    const int c4  = idx % (kKChunk / 4);
    goffs[j] = r * kInF + (c4 << 2);        // floats
    loffs[j] = r * kLda + (c4 << 2);        // floats
  }

#if HAS_ASYNC_LDS
  // Async DMA: global -> LDS directly (ASYNCcnt), no staging VGPRs.
  auto issue_chunk = [&](int c, int bufsel) {
    const float* src = xblk + (size_t)c * kKChunk;
    LDS_AS float* dstb = (LDS_AS float*)&As[bufsel][0];
    #pragma unroll
    for (int j = 0; j < 8; ++j) {
      if (j < 7 || tid < kV4Tail) {
        __builtin_amdgcn_global_load_async_to_lds_b128(
            (GLOBAL_AS v4i*)(src + goffs[j]),
            (LDS_AS v4i*)(dstb + loffs[j]),
            /*offset=*/0, /*cpol=*/0);
      }
    }
  };

  issue_chunk(0, 0);
  wait_async_zero();
  __syncthreads();
#else
  // Fallback: global -> regs -> LDS double buffering.
  float4 stg[8];
  auto load_chunk = [&](int c) {
    const float* src = xblk + (size_t)c * kKChunk;
    #pragma unroll
    for (int j = 0; j < 8; ++j)
      if (j < 7 || tid < kV4Tail) stg[j] = *(const float4*)(src + goffs[j]);
  };
  auto store_chunk = [&](int bufsel) {
    float* dst = As[bufsel];
    #pragma unroll
    for (int j = 0; j < 8; ++j)
      if (j < 7 || tid < kV4Tail) *(float4*)(dst + loffs[j]) = stg[j];
  };

  load_chunk(0);
  store_chunk(0);
  __syncthreads();
#endif

  for (int c = 0; c < kNChunks; ++c) {
    if (c + 1 < kNChunks) {
#if HAS_ASYNC_LDS
      issue_chunk(c + 1, (c + 1) & 1);                     // DMA next tile while we do WMMA
#else
      load_chunk(c + 1);
#endif
      __builtin_prefetch(wrow + (size_t)(c + 1) * kKChunk, 0, 1);  // global_prefetch_b8
    }

    const float* a0 = &As[c & 1][n * kLda + koff];
    const float* a1 = a0 + 16 * kLda;
    const float* wp = wrow + (size_t)c * kKChunk;
    #pragma unroll 6
    for (int kk = 0; kk < kKChunk / 4; ++kk) {
      v2f bf  = *(const v2f*)(wp + (kk << 2));             // B frag: global b64 (L2)
      v2f af0 = *(const v2f*)(a0 + (kk << 2));             // A frags: ds b64, conflict-free
      v2f af1 = *(const v2f*)(a1 + (kk << 2));
      acc0 = __builtin_amdgcn_wmma_f32_16x16x4_f32(false, af0, false, bf,
                                                   (short)0, acc0, false, false);
      acc1 = __builtin_amdgcn_wmma_f32_16x16x4_f32(false, af1, false, bf,
                                                   (short)0, acc1, false, false);
    }

    if (c + 1 < kNChunks) {
#if HAS_ASYNC_LDS
      wait_async_zero();                                   // this wave's DMA into buf nxt done
#else
      store_chunk((c + 1) & 1);
#endif
    }
    __syncthreads();
  }

  // ---- fused epilogue: bias + hardtanh, softmax over 16 classes, Gini impurity ----
  // C/D layout: VGPR v <-> M = v (lanes 0-15) / v+8 (lanes 16-31), N = lane&15
  const int   l      = wv * 16 + n;
  const float bb     = bias[l];
  const int   rowsel = (lane >> 4) << 3;    // 0 or 8
  float* outS = out;
  float* outG = out + (size_t)kM * kLeaves;

  const v8f accs[2] = {acc0, acc1};
  #pragma unroll
  for (int mt = 0; mt < kMTiles; ++mt) {
    const v8f a = accs[mt];
    #pragma unroll
    for (int v = 0; v < 8; ++v) {
      const int m = mbase + mt * 16 + rowsel + v;          // row in [B*T]
      float s = fminf(fmaxf(a[v] + bb, -1.0f), 1.0f);
      const int t = m & (kTrees - 1);                      // tree index = m % 64
      const float* cp = con + ((size_t)t * kLeaves + l) * kClasses;
      float4 c0 = *(const float4*)(cp + 0);
      float4 c1 = *(const float4*)(cp + 4);
      float4 c2 = *(const float4*)(cp + 8);
      float4 c3 = *(const float4*)(cp + 12);
      float z[kClasses];
      z[0]=s*c0.x;  z[1]=s*c0.y;  z[2]=s*c0.z;  z[3]=s*c0.w;
      z[4]=s*c1.x;  z[5]=s*c1.y;  z[6]=s*c1.z;  z[7]=s*c1.w;
      z[8]=s*c2.x;  z[9]=s*c2.y;  z[10]=s*c2.z; z[11]=s*c2.w;
      z[12]=s*c3.x; z[13]=s*c3.y; z[14]=s*c3.z; z[15]=s*c3.w;
      float mx = z[0];
      #pragma unroll
      for (int i = 1; i < kClasses; ++i) mx = fmaxf(mx, z[i]);
      float se = 0.0f, se2 = 0.0f;
      #pragma unroll
      for (int i = 0; i < kClasses; ++i) {
        float e = __expf(z[i] - mx);                       // v_exp_f32
        se  += e;
        se2 += e * e;
      }
      const float g = (float)kClasses - se2 / (se * se);   // sum(1 - p^2)
      const size_t o = (size_t)m * kLeaves + l;
      outS[o] = s;                                         // symbolic_paths
      outG[o] = g;                                         // class_value
    }
  }
}

extern "C" void kernel_launch(void* const* d_in, const int* in_sizes, int n_in,
                              void* d_out, int out_size, void* d_ws, size_t ws_size,
                              hipStream_t stream) {
  (void)in_sizes; (void)n_in; (void)out_size; (void)d_ws; (void)ws_size;
  const float* x   = (const float*)d_in[0];   // [64, 64, 16080]
  const float* Wm  = (const float*)d_in[1];   // [128, 16080]
  const float* b   = (const float*)d_in[2];   // [128]
  const float* con = (const float*)d_in[3];   // [64, 128, 16]
  float* out = (float*)d_out;                 // [2 * 4096 * 128]
  dim3 grid(kM / kMpb);                       // 128 blocks
  forest_fused_kernel<<<grid, kThreads, 0, stream>>>(x, Wm, b, con, out);
}